// BilateralDenoiser_44427141710592
// MI455X (gfx1250) — compile-verified
//
#include <hip/hip_runtime.h>
#include <cstdint>
#include <cstddef>

// ---------------------------------------------------------------------------
// Bilateral denoiser, MI455X (gfx1250).
// Input : (1, 1080, 1920, 8) f32  [col.rgb, n.xyz, z, dz]
// Output: (1, 1080, 1920, 3) f32
// ---------------------------------------------------------------------------

#define RAD    5
#define IMG_H  1080
#define IMG_W  1920
#define TW     32            // output tile width
#define TH     16            // output tile height
#define HLW    (TW + 2*RAD)  // 42 halo width
#define HLH    (TH + 2*RAD)  // 26 halo height
#define HPX    (HLW * HLH)   // 1092 halo pixels
#define NTHR   512           // 16 wave32s
#define EPSW   1.0e-4f

static_assert(HPX * 2 * 16 + 7 * HPX * 4 <= 65536, "LDS budget");

// ---- compile-time math (folded to literals; no runtime table loads) -------
constexpr double cexp_d(double x) {          // x in [-6.25, 0]
  double t = 1.0, s = 1.0;
  for (int i = 1; i < 64; ++i) { t *= x / i; s += t; }
  return s;
}
constexpr double csqrt_d(double x) {
  if (x <= 0.0) return 0.0;
  double r = (x > 1.0) ? x : 1.0;
  for (int i = 0; i < 64; ++i) r = 0.5 * (r + x / r);
  return r;
}

// ---- CDNA5 async global->LDS copy (ASYNCcnt path) -------------------------
typedef int v4i __attribute__((vector_size(16)));
typedef __attribute__((address_space(1))) v4i* gv4i_p;
typedef __attribute__((address_space(3))) v4i* lv4i_p;

__device__ __forceinline__ void async_ld16(const float* g, float4* l) {
#if __has_builtin(__builtin_amdgcn_global_load_async_to_lds_b128)
  __builtin_amdgcn_global_load_async_to_lds_b128(
      (gv4i_p)(g), (lv4i_p)(l), 0, 0);
#else
  unsigned lofs = (unsigned)(uintptr_t)(__attribute__((address_space(3))) void*)l;
  unsigned long long ga = (unsigned long long)(uintptr_t)g;
  asm volatile("global_load_async_to_lds_b128 %0, %1, off"
               :: "v"(lofs), "v"(ga) : "memory");
#endif
}
__device__ __forceinline__ void wait_async0() {
#if __has_builtin(__builtin_amdgcn_s_wait_asynccnt)
  __builtin_amdgcn_s_wait_asynccnt(0);
#else
  asm volatile("s_wait_asynccnt 0" ::: "memory");
#endif
}

__global__ __launch_bounds__(NTHR)
void bilateral_denoise_kernel(const float* __restrict__ in,
                              float* __restrict__ out) {
  // AoS staging (filled by async DMA), then SoA for conflict-free taps.
  __shared__ float4 s_stage[HPX * 2];          // 34944 B
  __shared__ float  s_soa[7][HPX];             // 30576 B: r,g,b,nx,ny,nz,z

  const int tid = threadIdx.x;
  const int x0  = blockIdx.x * TW;
  const int y0  = blockIdx.y * TH;

  // ---- phase 1: async-stage halo (32 B/pixel) into LDS --------------------
  for (int i = tid; i < HPX; i += NTHR) {
    const int hy = i / HLW;
    const int hx = i - hy * HLW;
    const int gy = y0 + hy - RAD;
    const int gx = x0 + hx - RAD;
    if ((unsigned)gx < (unsigned)IMG_W && (unsigned)gy < (unsigned)IMG_H) {
      const float* src = in + ((size_t)gy * IMG_W + gx) * 8;
      async_ld16(src,     &s_stage[2 * i]);
      async_ld16(src + 4, &s_stage[2 * i + 1]);
    } else {
      // zero-padded region: nrm==0 -> w_n==0 -> weight==0 (matches t_m==0)
      s_stage[2 * i]     = make_float4(0.f, 0.f, 0.f, 0.f);
      s_stage[2 * i + 1] = make_float4(0.f, 0.f, 0.f, 0.f);
    }
  }
  wait_async0();
  __syncthreads();

  // ---- phase 2: AoS -> SoA, normalize normals once per halo pixel ---------
  for (int i = tid; i < HPX; i += NTHR) {
    const float4 a = s_stage[2 * i];
    const float4 b = s_stage[2 * i + 1];
    const float nx = a.w, ny = b.x, nz = b.y;
    const float nn = nx * nx + ny * ny + nz * nz;
    const float inv = 1.0f / sqrtf(fmaxf(nn, 1.0e-20f));
    s_soa[0][i] = a.x;        // col.r
    s_soa[1][i] = a.y;        // col.g
    s_soa[2][i] = a.z;        // col.b
    s_soa[3][i] = nx * inv;   // nrm.x
    s_soa[4][i] = ny * inv;   // nrm.y
    s_soa[5][i] = nz * inv;   // nrm.z
    s_soa[6][i] = b.z;        // z
  }
  __syncthreads();

  // ---- phase 3: 121-tap bilateral accumulation ----------------------------
  const int tx = tid & (TW - 1);
  const int ty = tid >> 5;
  const int ci = (ty + RAD) * HLW + (tx + RAD);

  const float cnx = s_soa[3][ci];
  const float cny = s_soa[4][ci];
  const float cnz = s_soa[5][ci];
  const float cz  = s_soa[6][ci];
  const float cdz = s_stage[2 * ci + 1].w;     // dz only needed at center

  float ar = 0.f, ag = 0.f, ab = 0.f, aw = 0.f;

  auto tap = [&](int off, float wxy, float sd) {
    const int ti = ci + off;
    const float tnx = s_soa[3][ti];
    const float tny = s_soa[4][ti];
    const float tnz = s_soa[5][ti];
    float d = tnx * cnx + tny * cny + tnz * cnz;
    float wn = fminf(fmaxf(d, 0.0f), 1.0f);
    wn *= wn; wn *= wn; wn *= wn; wn *= wn;    // ^16
    wn *= wn; wn *= wn; wn *= wn;              // ^128
    const float tz  = s_soa[6][ti];
    const float ad  = fabsf(tz - cz);
    const float den = fmaxf(cdz * sd, EPSW);
    // exp(-ad/den) == exp2(-log2(e) * ad * rcp(den))
    const float wd =
        __builtin_amdgcn_exp2f(-1.4426950408889634f * ad *
                               __builtin_amdgcn_rcpf(den));
    const float w = wxy * wn * wd;
    ar += s_soa[0][ti] * w;
    ag += s_soa[1][ti] * w;
    ab += s_soa[2][ti] * w;
    aw += w;
  };

  // Fully unrolled 11x11 with constexpr-folded spatial weights: zero table
  // loads, DS offsets become immediates, adjacent taps vectorize.
#define TAP(DY, DX)                                                          \
  {                                                                          \
    constexpr float wxy_c =                                                  \
        (float)cexp_d(-(double)((DY) * (DY) + (DX) * (DX)) / 8.0);           \
    constexpr float sd_c =                                                   \
        (float)csqrt_d((double)((DY) * (DY) + (DX) * (DX)));                 \
    tap((DY) * HLW + (DX), wxy_c, sd_c);                                     \
  }
#define ROW(DY)                                                              \
  TAP(DY, -5) TAP(DY, -4) TAP(DY, -3) TAP(DY, -2) TAP(DY, -1) TAP(DY, 0)     \
  TAP(DY, 1) TAP(DY, 2) TAP(DY, 3) TAP(DY, 4) TAP(DY, 5)

  ROW(-5) ROW(-4) ROW(-3) ROW(-2) ROW(-1) ROW(0)
  ROW(1) ROW(2) ROW(3) ROW(4) ROW(5)

#undef ROW
#undef TAP

  const int gx = x0 + tx;
  const int gy = y0 + ty;
  if (gy < IMG_H) {
    const float invw = 1.0f / aw;              // center weight==1 -> aw>=1
    const size_t o = ((size_t)gy * IMG_W + gx) * 3;
    out[o + 0] = ar * invw;
    out[o + 1] = ag * invw;
    out[o + 2] = ab * invw;
  }
}

extern "C" void kernel_launch(void* const* d_in, const int* in_sizes, int n_in,
                              void* d_out, int out_size, void* d_ws,
                              size_t ws_size, hipStream_t stream) {
  (void)in_sizes; (void)n_in; (void)out_size; (void)d_ws; (void)ws_size;
  const float* in = (const float*)d_in[0];
  float* out = (float*)d_out;
  dim3 grid((IMG_W + TW - 1) / TW, (IMG_H + TH - 1) / TH, 1);  // 60 x 68
  dim3 block(NTHR, 1, 1);
  bilateral_denoise_kernel<<<grid, block, 0, stream>>>(in, out);
}